// DecoderCRF_85504208928910
// MI455X (gfx1250) — compile-verified
//
#include <hip/hip_runtime.h>
#include <hip/hip_bf16.h>

typedef __attribute__((ext_vector_type(2))) float v2f;
typedef __attribute__((ext_vector_type(8))) float v8f;

#define T_TAGS    12
#define NTPAD     16
#define BB        64
#define SS        512
#define DD        512
#define START_TAG 10
#define STOP_TAG  11
#define NEGV      (-10000.0f)
#define WSTRIDE   516   // LDS row stride (floats): 4m mod 64 distinct -> no bank conflicts

// ---------------------------------------------------------------------------
// Kernel 1: feats[b,s,t] = sum_d input[b,s,d] * W[t,d] + bias[t]
// One wave = one 16x16 output tile (16 flat (b,s)-rows x 16 tag-cols, T padded
// 12->16). 128 x V_WMMA_F32_16X16X4_F32 over K=512, two accumulator chains.
// A 16x4 layout (ISA 7.12.2): VGPR0 = {K=0 | K=2}, VGPR1 = {K=1 | K=3},
// M = lane%16.  B 4x16 mirrored with N = lane%16.  B staged in LDS (zero-
// padded rows 12..15) so the inner loop has no EXEC-divergent branches.
// ---------------------------------------------------------------------------
__global__ __launch_bounds__(256) void crf_feats_wmma(
    const float* __restrict__ in, const float* __restrict__ W,
    const float* __restrict__ bias, float* __restrict__ feats)
{
  __shared__ float wlds[16 * WSTRIDE];

  const int lane  = threadIdx.x & 31;
  const int wave  = threadIdx.x >> 5;
  const int tile  = blockIdx.x * 8 + wave;     // 2048 tiles of 16 rows
  const int m     = lane & 15;                 // A-row-in-tile == B-column
  const int khalf = (lane >> 4) << 1;          // 0 or 2

  // Stage W (12x512) into LDS, rows 12..15 zeroed. Coalesced b32 loads.
  for (int idx = threadIdx.x; idx < 16 * DD; idx += 256) {
    const int r = idx >> 9;           // 0..15
    const int ccol = idx & (DD - 1);  // 0..511
    wlds[r * WSTRIDE + ccol] = (r < T_TAGS) ? W[(r << 9) + ccol] : 0.0f;
  }
  __syncthreads();

  const int    row  = tile * 16 + m;           // flat b*S+s row
  const float* arow = in + (size_t)row * DD;
  const float* wrow = &wlds[m * WSTRIDE];
  const float  bn   = (m < T_TAGS) ? bias[m] : 0.0f;

  v8f c0 = {bn, bn, bn, bn, bn, bn, bn, bn};   // bias folded into accumulator
  v8f c1 = {};

  for (int k0 = 0; k0 < DD; k0 += 64) {
    // Prefetch the A stream ~1KB ahead (speculative; OOB silently dropped).
    __builtin_prefetch(arow + k0 + 256, 0, 3);
#pragma unroll
    for (int kk = 0; kk < 64; kk += 8) {
      const int col = k0 + kk + khalf;
      v2f a0 = *reinterpret_cast<const v2f*>(arow + col);
      v2f b0 = *reinterpret_cast<const v2f*>(wrow + col);
      v2f a1 = *reinterpret_cast<const v2f*>(arow + col + 4);
      v2f b1 = *reinterpret_cast<const v2f*>(wrow + col + 4);
      c0 = __builtin_amdgcn_wmma_f32_16x16x4_f32(false, a0, false, b0,
                                                 (short)0, c0, false, false);
      c1 = __builtin_amdgcn_wmma_f32_16x16x4_f32(false, a1, false, b1,
                                                 (short)0, c1, false, false);
    }
  }

  // C/D layout: VGPR i holds rows {i | i+8}, N = lane%16
  const int rbase = tile * 16 + ((lane >> 4) << 3);
#pragma unroll
  for (int i = 0; i < 8; ++i)
    feats[(size_t)(rbase + i) * NTPAD + m] = c0[i] + c1[i];
}

// ---------------------------------------------------------------------------
// Kernel 2: per-batch CRF forward scan + gold score. One wave32 per batch.
// Lanes 0..11 each own destination tag j; fv[12] replicated in every lane.
// ---------------------------------------------------------------------------
__global__ __launch_bounds__(32) void crf_scan(
    const float* __restrict__ feats, const float* __restrict__ trans,
    const int* __restrict__ tags, const int* __restrict__ mask,
    float* __restrict__ partial)
{
  const int b    = blockIdx.x;
  const int lane = threadIdx.x;
  const int j    = (lane < T_TAGS) ? lane : 0;

  float tr[T_TAGS];
#pragma unroll
  for (int i = 0; i < T_TAGS; ++i) tr[i] = trans[j * T_TAGS + i];

  float fvv[T_TAGS];
#pragma unroll
  for (int i = 0; i < T_TAGS; ++i) fvv[i] = (i == START_TAG) ? 0.0f : NEGV;

  const float* frow = feats + (size_t)b * SS * NTPAD;
  const int*   mrow = mask + b * SS;
  const int*   trow = tags + b * SS;

  for (int s = 0; s < SS; ++s) {
    const float ft = frow[s * NTPAD + j];
    const float mt = (float)mrow[s];

    float mmax = fvv[0] + tr[0];
#pragma unroll
    for (int i = 1; i < T_TAGS; ++i) mmax = fmaxf(mmax, fvv[i] + tr[i]);
    float ssum = 0.0f;
#pragma unroll
    for (int i = 0; i < T_TAGS; ++i) ssum += __expf(fvv[i] + tr[i] - mmax);
    float newv = ft + mmax + __logf(ssum);
    newv = mt * newv + (1.0f - mt) * fvv[j];
#pragma unroll
    for (int i = 0; i < T_TAGS; ++i) fvv[i] = __shfl(newv, i, 32);
  }

  // alpha = logsumexp(fv + transitions[STOP,:]) — replicated across lanes
  float am = fvv[0] + trans[STOP_TAG * T_TAGS + 0];
#pragma unroll
  for (int i = 1; i < T_TAGS; ++i)
    am = fmaxf(am, fvv[i] + trans[STOP_TAG * T_TAGS + i]);
  float asum = 0.0f;
#pragma unroll
  for (int i = 0; i < T_TAGS; ++i)
    asum += __expf(fvv[i] + trans[STOP_TAG * T_TAGS + i] - am);
  const float alpha = am + __logf(asum);

  // gold score: lane-strided over S, shfl-xor reduction
  float esum = 0.0f, tsum = 0.0f;
  int   msum = 0;
  for (int s = lane; s < SS; s += 32) {
    const int tg = trow[s];
    const int mk = mrow[s];
    msum += mk;
    if (s < SS - 1) {
      esum += frow[s * NTPAD + tg] * (float)mk;
      tsum += trans[trow[s + 1] * T_TAGS + tg] * (float)mrow[s + 1];
    }
  }
#pragma unroll
  for (int off = 16; off > 0; off >>= 1) {
    esum += __shfl_xor(esum, off, 32);
    tsum += __shfl_xor(tsum, off, 32);
    msum += __shfl_xor(msum, off, 32);
  }

  if (lane == 0) {
    const int   last_idx = msum - 1;
    const int   lastt    = trow[last_idx];
    const float score0   = trans[trow[0] * T_TAGS + START_TAG];
    const float lastemit = frow[(SS - 1) * NTPAD + lastt] * (float)mrow[SS - 1];
    const float gold = score0 + esum + tsum +
                       trans[STOP_TAG * T_TAGS + lastt] + lastemit;
    partial[b] = alpha - gold;
  }
}

// ---------------------------------------------------------------------------
// Kernel 3: deterministic 64 -> 1 reduction
// ---------------------------------------------------------------------------
__global__ __launch_bounds__(32) void crf_reduce(
    const float* __restrict__ partial, float* __restrict__ out)
{
  const int lane = threadIdx.x;
  float v = partial[lane] + partial[lane + 32];
#pragma unroll
  for (int off = 16; off > 0; off >>= 1) v += __shfl_xor(v, off, 32);
  if (lane == 0) out[0] = v;
}

extern "C" void kernel_launch(void* const* d_in, const int* in_sizes, int n_in,
                              void* d_out, int out_size, void* d_ws, size_t ws_size,
                              hipStream_t stream) {
  const float* in    = (const float*)d_in[0];   // (64,512,512) f32
  const float* W     = (const float*)d_in[1];   // (12,512) f32
  const float* bias  = (const float*)d_in[2];   // (12,) f32
  const float* trans = (const float*)d_in[3];   // (12,12) f32
  const int*   tags  = (const int*)d_in[4];     // (64,512) i32
  const int*   mask  = (const int*)d_in[5];     // (64,512) i32

  float* feats   = (float*)d_ws;                       // 32768*16 f32 = 2 MB
  float* partial = feats + (size_t)BB * SS * NTPAD;    // 64 f32

  crf_feats_wmma<<<dim3(256), dim3(256), 0, stream>>>(in, W, bias, feats);
  crf_scan<<<dim3(BB), dim3(32), 0, stream>>>(feats, trans, tags, mask, partial);
  crf_reduce<<<dim3(1), dim3(32), 0, stream>>>(partial, (float*)d_out);
}